// DGCNNLayer_2044404433240
// MI455X (gfx1250) — compile-verified
//
#include <hip/hip_runtime.h>

// ---------------------------------------------------------------------------
// DGCNN edge-conv layer for MI455X (gfx1250), wave32 + WMMA f32 16x16x4 +
// async global->LDS staging (ASYNCcnt) with double buffering.
//
// Shapes (fixed by reference): B=4, N=8192, C=64, O=64, K=20.
// d_in[0]=x (B,N,C) f32; d_in[1]=W1 (128,64) f32; d_in[2]=W2 (64,64) f32.
// d_out = [ out (B,N,64) f32 | idx (B,N,20) i32 ] flat.
// d_ws  = [ xx (B*N) f32 | c1 (B*N,64) f32 ]  (~8.5 MB used).
// ---------------------------------------------------------------------------

#define BB 4
#define NN 8192
#define CC 64
#define OO 64
#define KK 20
#define SLOPE 0.2f

typedef __attribute__((ext_vector_type(2))) float v2f;
typedef __attribute__((ext_vector_type(8))) float v8f;

// D = A(16x4 f32) x B(4x16 f32) + C(16x16 f32), wave32 layouts per ISA 7.12.2.
__device__ __forceinline__ v8f wmma4(v2f a, v2f b, v8f c) {
  return __builtin_amdgcn_wmma_f32_16x16x4_f32(
      /*neg_a=*/false, a, /*neg_b=*/false, b,
      /*c_mod=*/(short)0, c, /*reuse_a=*/false, /*reuse_b=*/false);
}

__device__ __forceinline__ float leaky(float v) { return v > 0.0f ? v : SLOPE * v; }

// Order-preserving float<->uint map so K-max can use LDS atomicMax(u32).
__device__ __forceinline__ unsigned ford(float f) {
  unsigned u = __float_as_uint(f);
  return (u & 0x80000000u) ? ~u : (u | 0x80000000u);
}
__device__ __forceinline__ float funord(unsigned u) {
  return __uint_as_float((u & 0x80000000u) ? (u & 0x7fffffffu) : ~u);
}

// Generic LDS pointer -> workgroup-relative byte offset (flat aperture rule:
// the LDS offset is the low 32 bits of the address).
__device__ __forceinline__ unsigned lds_off(const void* p) {
  return (unsigned)(uintptr_t)p;
}

// Async 16-byte global->LDS copy, tracked by ASYNCcnt (gfx1250).
__device__ __forceinline__ void async_g2l_b128(unsigned lds_byte_off,
                                               const float* gaddr) {
  asm volatile("global_load_async_to_lds_b128 %0, %1, off"
               :
               : "v"(lds_byte_off), "v"(gaddr)
               : "memory");
}

__device__ __forceinline__ void wait_async0() {
#if __has_builtin(__builtin_amdgcn_s_wait_asynccnt)
  __builtin_amdgcn_s_wait_asynccnt(0);
#else
  asm volatile("s_wait_asynccnt 0x0" ::: "memory");
#endif
}

// ---------------------------------------------------------------------------
// Kernel 1: squared norms xx[b*N+n] = sum_c x^2
// ---------------------------------------------------------------------------
__global__ void __launch_bounds__(256) norms_k(const float* __restrict__ x,
                                               float* __restrict__ xx) {
  int i = blockIdx.x * 256 + threadIdx.x;          // point id, 32768 total
  const float4* p = (const float4*)(x + (size_t)i * CC);
  float s = 0.0f;
#pragma unroll
  for (int q = 0; q < CC / 4; ++q) {
    float4 v = p[q];
    s += v.x * v.x + v.y * v.y + v.z * v.z + v.w * v.w;
  }
  xx[i] = s;
}

// ---------------------------------------------------------------------------
// Kernel 2: c1 = x @ (W1b - W1a), the per-point "center" term of edge@W1.
// ---------------------------------------------------------------------------
__global__ void __launch_bounds__(256) c1_k(const float* __restrict__ x,
                                            const float* __restrict__ W1,
                                            float* __restrict__ c1) {
  __shared__ float Wd[CC * OO];                    // W1[64:128]-W1[0:64], 16 KB
  int t = threadIdx.x;
  for (int i = t; i < CC * OO; i += 256) Wd[i] = W1[CC * OO + i] - W1[i];
  __syncthreads();

  int wave = t >> 5, lane = t & 31, lo = lane & 15, hi = lane >> 4;
  int mt = blockIdx.x * 8 + wave;                  // M-tile id, 2048 total
  const float* ar = x + ((size_t)mt * 16 + lo) * CC;

  v2f av[16];
#pragma unroll
  for (int s = 0; s < 16; ++s) {
    av[s].x = ar[4 * s + 2 * hi];
    av[s].y = ar[4 * s + 2 * hi + 1];
  }
#pragma unroll
  for (int nt = 0; nt < 4; ++nt) {
    v8f acc = {};
#pragma unroll
    for (int s = 0; s < 16; ++s) {
      v2f bo;
      bo.x = Wd[(4 * s + 2 * hi) * OO + nt * 16 + lo];
      bo.y = Wd[(4 * s + 2 * hi + 1) * OO + nt * 16 + lo];
      acc = wmma4(av[s], bo, acc);
    }
#pragma unroll
    for (int r = 0; r < 8; ++r) {
      int M = hi * 8 + r;
      c1[((size_t)mt * 16 + M) * OO + nt * 16 + lo] = acc[r];
    }
  }
}

// ---------------------------------------------------------------------------
// Kernel 3: fused Gram-matrix + top-K.  Workgroup = (batch b, 16 rows).
// Column chunks of 128 are streamed global->LDS with async b128 copies,
// double-buffered against the WMMA sweep; 8 waves each produce one 16x16
// distance tile -> LDS; 16 threads/row keep register top-20 lists; final
// 320 -> 20 merge per row.  Self-point included (as in reference).
// ---------------------------------------------------------------------------
__global__ void __launch_bounds__(256) knn_k(const float* __restrict__ x,
                                             const float* __restrict__ xx,
                                             int* __restrict__ idx_out) {
  __shared__ float As[16 * CC];                    // 4 KB  row features
  __shared__ float rowN[16];
  __shared__ float dist[16][128];                  // 8 KB  distance chunk
  __shared__ float mv[16][16 * KK];                // 20 KB merge values
  __shared__ int mi[16][16 * KK];                  // 20 KB merge indices
  __shared__ float bstage[2][128 * CC];            // 64 KB double-buffered B

  int b = blockIdx.y;
  int row0 = blockIdx.x * 16;
  int t = threadIdx.x;
  const float* xb = x + (size_t)b * NN * CC;
  const float* xxb = xx + (size_t)b * NN;

  // stage chunk [j0, j0+128) of columns: 32 KB, 2048 b128 ops, 8 per thread
  auto stage = [&](int j0, int pb) {
    const float* g = xb + (size_t)j0 * CC;
    unsigned base = lds_off(&bstage[pb][0]);
#pragma unroll
    for (int q = 0; q < 8; ++q) {
      int e = t + 256 * q;                         // b128 transfer id
      async_g2l_b128(base + e * 16, g + e * 4);
    }
  };

  stage(0, 0);                                     // prologue chunk 0
  for (int i = t; i < 16 * CC; i += 256) As[i] = xb[(size_t)row0 * CC + i];
  if (t < 16) rowN[t] = xxb[row0 + t];
  wait_async0();
  __syncthreads();

  int wave = t >> 5, lane = t & 31, lo = lane & 15, hi = lane >> 4;

  v2f av[16];                                      // this lane's A-row slices
#pragma unroll
  for (int s = 0; s < 16; ++s) {
    av[s].x = As[lo * CC + 4 * s + 2 * hi];
    av[s].y = As[lo * CC + 4 * s + 2 * hi + 1];
  }

  float tv[KK];
  int ti[KK];
#pragma unroll
  for (int s = 0; s < KK; ++s) { tv[s] = -3.4e38f; ti[s] = 0; }

  int selRow = t >> 4;                             // 0..15
  int selSlot = t & 15;                            // 0..15

  for (int j0 = 0; j0 < NN; j0 += 128) {
    int pb = (j0 >> 7) & 1;
    if (j0 + 128 < NN) stage(j0 + 128, pb ^ 1);    // uniform branch

    const float* bcol = &bstage[pb][(wave * 16 + lo) * CC];
    float cn = xxb[j0 + wave * 16 + lo];

    v8f acc = {};
#pragma unroll
    for (int s = 0; s < 16; ++s) {
      v2f bo;
      bo.x = bcol[4 * s + 2 * hi];
      bo.y = bcol[4 * s + 2 * hi + 1];
      acc = wmma4(av[s], bo, acc);
    }
#pragma unroll
    for (int r = 0; r < 8; ++r) {
      int M = hi * 8 + r;
      dist[M][wave * 16 + lo] = 2.0f * acc[r] - rowN[M] - cn;
    }
    __syncthreads();                               // publish dist

    // selection: thread (row, slot) scans 8 columns of the chunk
#pragma unroll
    for (int q = 0; q < 8; ++q) {
      int cc2 = selSlot * 8 + q;
      float d = dist[selRow][cc2];
      int j = j0 + cc2;
      if (d > tv[KK - 1]) {
        tv[KK - 1] = d;
        ti[KK - 1] = j;
#pragma unroll
        for (int s2 = KK - 1; s2 > 0; --s2) {
          if (tv[s2] > tv[s2 - 1]) {
            float fv = tv[s2]; tv[s2] = tv[s2 - 1]; tv[s2 - 1] = fv;
            int iv = ti[s2]; ti[s2] = ti[s2 - 1]; ti[s2 - 1] = iv;
          }
        }
      }
    }
    wait_async0();                                 // next chunk landed (own wave)
    __syncthreads();                               // all waves done; dist reusable
  }

  // merge 16 sorted lists of 20 per row -> final descending top-20
#pragma unroll
  for (int s = 0; s < KK; ++s) {
    mv[selRow][selSlot * KK + s] = tv[s];
    mi[selRow][selSlot * KK + s] = ti[s];
  }
  __syncthreads();
  if (t < 16) {
    int row = t;
    for (int s = 0; s < KK; ++s) {
      float best = -3.4e38f;
      int bi = 0, bp = 0;
      for (int i = 0; i < 16 * KK; ++i) {
        float v = mv[row][i];
        if (v > best) { best = v; bi = mi[row][i]; bp = i; }
      }
      mv[row][bp] = -3.4e38f;
      idx_out[((size_t)b * NN + row0 + row) * KK + s] = bi;
    }
  }
}

// ---------------------------------------------------------------------------
// Kernel 4: fused gather + MLP + K-max.  Each wave owns 4 points = exactly
// 80 edge rows = five 16-row WMMA M-tiles (no padding).
//   h1 = leaky( neigh@W1a + c1[point] )  -> per-wave LDS (C/D->A relayout)
//   h2 = leaky( h1@W2 ); out = max_k h2  via order-mapped LDS atomicMax.
// W1/W2 are staged into LDS with async b128 copies.
// ---------------------------------------------------------------------------
__global__ void __launch_bounds__(256) edge_k(const float* __restrict__ x,
                                              const float* __restrict__ W1,
                                              const float* __restrict__ W2,
                                              const float* __restrict__ c1,
                                              const int* __restrict__ idx_in,
                                              float* __restrict__ out) {
  __shared__ float W1s[CC * OO];                   // W1a, 16 KB
  __shared__ float W2s[CC * OO];                   // 16 KB
  __shared__ float h1s[8][16 * OO];                // per-wave h1 tile, 32 KB
  __shared__ float c1s[8][4 * OO];                 // per-wave bias, 8 KB
  __shared__ unsigned omax[8][4 * OO];             // per-wave K-max, 8 KB

  int t = threadIdx.x;
  {
    unsigned b1 = lds_off(W1s), b2 = lds_off(W2s); // 1024 b128 ops each
#pragma unroll
    for (int q = 0; q < 4; ++q) {
      int e = t + 256 * q;
      async_g2l_b128(b1 + e * 16, W1 + e * 4);
      async_g2l_b128(b2 + e * 16, W2 + e * 4);
    }
  }
  int wave = t >> 5, lane = t & 31, lo = lane & 15, hi = lane >> 4;
  int gp0 = blockIdx.x * 32 + wave * 4;            // first of this wave's 4 points
  __builtin_prefetch(idx_in + (size_t)gp0 * KK, 0, 0);
#pragma unroll
  for (int q = 0; q < 8; ++q) {
    int e = lane + 32 * q;                         // 0..255
    c1s[wave][e] = c1[(size_t)gp0 * OO + e];       // 4 points x 64, contiguous
    omax[wave][e] = 0u;                            // ford(x) > 0 for all finite x
  }
  wait_async0();
  __syncthreads();                                 // publish W1s/W2s

  int b = gp0 / NN;                                // group never crosses batches
  const float* xb = x + (size_t)b * NN * CC;

  for (int mt = 0; mt < 5; ++mt) {
    // gather this lane's neighbor row for rows mt*16 .. mt*16+15
    int grow = mt * 16 + lo;                       // 0..79
    int p = grow / KK, k = grow % KK;
    int nb = idx_in[(size_t)(gp0 + p) * KK + k];
    const float* ar = xb + (size_t)nb * CC;
    v2f av[16];
#pragma unroll
    for (int s = 0; s < 16; ++s) {
      av[s].x = ar[4 * s + 2 * hi];
      av[s].y = ar[4 * s + 2 * hi + 1];
    }

    // stage 1: h1 = leaky(neigh @ W1a + c1)
#pragma unroll
    for (int nt = 0; nt < 4; ++nt) {
      v8f acc = {};
#pragma unroll
      for (int s = 0; s < 16; ++s) {
        v2f bo;
        bo.x = W1s[(4 * s + 2 * hi) * OO + nt * 16 + lo];
        bo.y = W1s[(4 * s + 2 * hi + 1) * OO + nt * 16 + lo];
        acc = wmma4(av[s], bo, acc);
      }
#pragma unroll
      for (int r = 0; r < 8; ++r) {
        int M = hi * 8 + r;
        int pp = (mt * 16 + M) / KK;
        int o = nt * 16 + lo;
        h1s[wave][M * OO + o] = leaky(acc[r] + c1s[wave][pp * OO + o]);
      }
    }

    // stage 2: h2 = leaky(h1 @ W2); fold into per-(point,channel) max
#pragma unroll
    for (int nt = 0; nt < 4; ++nt) {
      v8f acc = {};
#pragma unroll
      for (int s = 0; s < 16; ++s) {
        v2f a2, b2;
        a2.x = h1s[wave][lo * OO + 4 * s + 2 * hi];
        a2.y = h1s[wave][lo * OO + 4 * s + 2 * hi + 1];
        b2.x = W2s[(4 * s + 2 * hi) * OO + nt * 16 + lo];
        b2.y = W2s[(4 * s + 2 * hi + 1) * OO + nt * 16 + lo];
        acc = wmma4(a2, b2, acc);
      }
#pragma unroll
      for (int r = 0; r < 8; ++r) {
        int M = hi * 8 + r;
        int pp = (mt * 16 + M) / KK;
        int o = nt * 16 + lo;
        atomicMax(&omax[wave][pp * OO + o], ford(leaky(acc[r])));
      }
    }
  }

  // write this wave's 4 points (4 x 64 contiguous floats)
#pragma unroll
  for (int q = 0; q < 8; ++q) {
    int e = lane + 32 * q;
    out[(size_t)gp0 * OO + e] = funord(omax[wave][e]);
  }
}

// ---------------------------------------------------------------------------
extern "C" void kernel_launch(void* const* d_in, const int* in_sizes, int n_in,
                              void* d_out, int out_size, void* d_ws, size_t ws_size,
                              hipStream_t stream) {
  const float* x = (const float*)d_in[0];
  const float* W1 = (const float*)d_in[1];
  const float* W2 = (const float*)d_in[2];

  float* out = (float*)d_out;
  int* idx_out = (int*)(out + (size_t)BB * NN * OO);   // idx follows out, flat

  float* xx = (float*)d_ws;                            // B*N floats
  float* c1 = xx + (size_t)BB * NN;                    // B*N*64 floats (~8 MB)

  norms_k<<<dim3(BB * NN / 256), dim3(256), 0, stream>>>(x, xx);
  c1_k<<<dim3(BB * NN / (16 * 8)), dim3(256), 0, stream>>>(x, W1, c1);
  knn_k<<<dim3(NN / 16, BB), dim3(256), 0, stream>>>(x, xx, idx_out);
  edge_k<<<dim3(BB * NN / 32), dim3(256), 0, stream>>>(x, W1, W2, c1, idx_out, out);
}